// MoEMapper_23098334118398
// MI455X (gfx1250) — compile-verified
//
#include <hip/hip_runtime.h>

#define B_ROWS 2048
#define T_STEPS 8
#define IN_DIM 1024
#define OUT_DIM 4096
#define NEXP 12

typedef __attribute__((ext_vector_type(2))) float v2f;
typedef __attribute__((ext_vector_type(8))) float v8f;

// ---------------------------------------------------------------------------
// Kernel 0: zero the per-expert row counters.
// ---------------------------------------------------------------------------
__global__ void zero_cnt_kernel(int* __restrict__ cnt) {
    if (threadIdx.x < NEXP) cnt[threadIdx.x] = 0;
}

// ---------------------------------------------------------------------------
// Kernel 1: gate.  One block per row b.
//   t_mean = mean over T of t[b]  (fused, never materialized)
//   logits = t_mean @ Wg + bg ; top1 = argmax (first-max, like jnp.argmax)
//   append b to rowlist[top1]
// Streams 256 MB of t coalesced -> pure bandwidth (~11 us floor at 23.3 TB/s).
// ---------------------------------------------------------------------------
__global__ __launch_bounds__(256) void gate_kernel(
    const float* __restrict__ t,
    const float* __restrict__ Wg,
    const float* __restrict__ bg,
    int* __restrict__ cnt,
    int* __restrict__ rowlist)
{
    const int b   = blockIdx.x;
    const int tid = threadIdx.x;

    __shared__ float sacc[NEXP];
    if (tid < NEXP) sacc[tid] = 0.0f;
    __syncthreads();

    float acc[NEXP];
#pragma unroll
    for (int e = 0; e < NEXP; ++e) acc[e] = 0.0f;

    const float* tb = t + (size_t)b * T_STEPS * OUT_DIM;
    for (int o = tid; o < OUT_DIM; o += 256) {
        float s = 0.0f;
#pragma unroll
        for (int tt = 0; tt < T_STEPS; ++tt) s += tb[tt * OUT_DIM + o];
        const float mean = s * (1.0f / (float)T_STEPS);
        const float* wg = Wg + o * NEXP;
#pragma unroll
        for (int e = 0; e < NEXP; ++e) acc[e] += mean * wg[e];
    }
#pragma unroll
    for (int e = 0; e < NEXP; ++e) atomicAdd(&sacc[e], acc[e]);  // ds_add_f32
    __syncthreads();

    if (tid == 0) {
        int best = 0;
        float bv = sacc[0] + bg[0];
#pragma unroll
        for (int e = 1; e < NEXP; ++e) {
            const float v = sacc[e] + bg[e];
            if (v > bv) { bv = v; best = e; }   // strict > keeps first max
        }
        const int pos = atomicAdd(&cnt[best], 1);
        rowlist[best * B_ROWS + pos] = b;
    }
}

// ---------------------------------------------------------------------------
// Kernel 2: routed GEMM via V_WMMA_F32_16X16X4_F32.
// Block = 128 threads (4 waves), block tile = 16 rows x 128 columns.
// Each wave computes a 16x32 sub-tile: TWO independent C fragments sharing
// one A fragment -> two interleaved WMMA accumulation chains (2x matrix-pipe
// ILP vs a single serial D->C chain), and half the A-side LDS traffic.
//   A (16x4 f32, 2 VGPR): lanes 0-15 = M, vgpr0/1 = K,K+1 (lanes 16-31: K+2,K+3)
//   B (4x16 f32, 2 VGPR): lanes 0-15 = N, vgpr0/1 = K,K+1 (lanes 16-31: K+2,K+3)
//   C (16x16 f32, 8 VGPR): lane = N, vgpr v = row v (lanes 16-31: row v+8)
// x tile staged in LDS via float4 (b128) copies, bank-padded stride 260
// (260 % 64 == 4 -> conflict-free 16x4 A gathers). W rows are read
// lane-coalesced; all of W (192 MB) is L2-resident so HBM sees it once.
// ---------------------------------------------------------------------------
#define KCHUNK     256
#define LDS_STRIDE (KCHUNK + 4)

__global__ __launch_bounds__(128) void moe_gemm_kernel(
    const float* __restrict__ x,
    const float* __restrict__ W,
    const float* __restrict__ bias,
    const int* __restrict__ cnt,
    const int* __restrict__ rowlist,
    float* __restrict__ out)
{
    // Map global m-tile (blockIdx.y) -> (expert e, local tile mloc) by
    // walking per-expert tile counts. Blocks past the total exit (uniform).
    int mloc = blockIdx.y;
    int e = 0, ce = 0;
    for (e = 0; e < NEXP; ++e) {
        ce = cnt[e];
        const int te = (ce + 15) >> 4;
        if (mloc < te) break;
        mloc -= te;
    }
    if (e >= NEXP) return;

    const int tid  = threadIdx.x;
    const int lane = tid & 31;
    const int wave = tid >> 5;
    const int l15  = lane & 15;
    const int half = lane >> 4;          // 0: lanes 0-15, 1: lanes 16-31
    const int koff = half << 1;          // K offset 0 or 2 within the 4-step
    const int n0   = blockIdx.x * 128 + wave * 32;   // wave owns [n0, n0+32)

    __shared__ float ldsX[16 * LDS_STRIDE];
    __shared__ int   srow[16];

    const int rbase = mloc << 4;
    if (tid < 16) {
        const int r = rbase + tid;
        srow[tid] = (r < ce) ? rowlist[e * B_ROWS + r] : -1;
    }
    __syncthreads();

    // Fixed staging assignment: 8 threads per row, float4 columns.
    const int fr    = tid >> 3;          // row 0..15
    const int fc    = (tid & 7) << 2;    // starting float column (0,4,..,28)
    const int myrow = srow[fr];
    const float* xr = (myrow >= 0) ? (x + (size_t)myrow * IN_DIM) : nullptr;

    const float* Wp = W + (size_t)e * IN_DIM * OUT_DIM + n0 + l15;
    v8f c0 = {};
    v8f c1 = {};

    for (int kc = 0; kc < IN_DIM; kc += KCHUNK) {
        // Stage x[rows, kc:kc+KCHUNK] into LDS: b128 load + b128 store,
        // 8 straight-line iterations, row test resolved once per thread.
#pragma unroll
        for (int i = 0; i < KCHUNK / 32; ++i) {
            const int col = fc + i * 32;
            float4 v = make_float4(0.f, 0.f, 0.f, 0.f);
            if (xr) v = *(const float4*)(xr + kc + col);
            *(float4*)&ldsX[fr * LDS_STRIDE + col] = v;
        }
        __syncthreads();

        const float* wk = Wp + (size_t)kc * OUT_DIM;
#pragma unroll 4
        for (int k = 0; k < KCHUNK; k += 4) {
            v2f a;
            const float* ap = &ldsX[l15 * LDS_STRIDE + k + koff];
            a.x = ap[0];
            a.y = ap[1];
            const float* w0 = wk + (size_t)(k + koff) * OUT_DIM;
            const float* w1 = wk + (size_t)(k + koff + 1) * OUT_DIM;
            v2f b0; b0.x = w0[0];  b0.y = w1[0];    // columns n0+l15
            v2f b1; b1.x = w0[16]; b1.y = w1[16];   // columns n0+16+l15
            c0 = __builtin_amdgcn_wmma_f32_16x16x4_f32(
                    false, a, false, b0, (short)0, c0, false, false);
            c1 = __builtin_amdgcn_wmma_f32_16x16x4_f32(
                    false, a, false, b1, (short)0, c1, false, false);
        }
        __syncthreads();
    }

    // Epilogue: bias + scatter per the C layout (lane = column, vgpr = row).
    const int colA = n0 + l15;
    const int colB = colA + 16;
    const float bA = bias[e * OUT_DIM + colA];
    const float bB = bias[e * OUT_DIM + colB];
#pragma unroll
    for (int v = 0; v < 8; ++v) {
        const int m   = v + half * 8;
        const int row = srow[m];
        if (row >= 0) {
            out[(size_t)row * OUT_DIM + colA] = c0[v] + bA;
            out[(size_t)row * OUT_DIM + colB] = c1[v] + bB;
        }
    }
}

// ---------------------------------------------------------------------------
extern "C" void kernel_launch(void* const* d_in, const int* in_sizes, int n_in,
                              void* d_out, int out_size, void* d_ws, size_t ws_size,
                              hipStream_t stream) {
    const float* x  = (const float*)d_in[0];   // (B, 1, IN_DIM)
    const float* t  = (const float*)d_in[1];   // (B, T, OUT_DIM)
    const float* W  = (const float*)d_in[2];   // (E, IN_DIM, OUT_DIM)
    const float* bb = (const float*)d_in[3];   // (E, OUT_DIM)
    const float* Wg = (const float*)d_in[4];   // (OUT_DIM, E)
    const float* bg = (const float*)d_in[5];   // (E,)
    float* out = (float*)d_out;                // (B, 1, OUT_DIM)

    int* cnt     = (int*)d_ws;                 // NEXP ints
    int* rowlist = (int*)((char*)d_ws + 256);  // NEXP * B ints

    zero_cnt_kernel<<<1, 32, 0, stream>>>(cnt);
    gate_kernel<<<B_ROWS, 256, 0, stream>>>(t, Wg, bg, cnt, rowlist);

    // Worst case M-tiles across experts: B/16 full tiles + one partial per expert.
    dim3 grid(OUT_DIM / 128, B_ROWS / 16 + NEXP);
    moe_gemm_kernel<<<grid, 128, 0, stream>>>(x, W, bb, cnt, rowlist, out);
}